// GConv_755914244835
// MI455X (gfx1250) — compile-verified
//
#include <hip/hip_runtime.h>

// ---------------------------------------------------------------------------
// 2-layer GCN for MI455X (gfx1250, wave32).
//   h   = H @ W                         (fp32 WMMA 16x16x4)
//   acc = b + sum_e norm(e) * h[row(e)] scattered to col(e), + self loop
//   out = relu(acc)
// ---------------------------------------------------------------------------

typedef float v2f __attribute__((ext_vector_type(2)));
typedef float v8f __attribute__((ext_vector_type(8)));

// ---------------- degree / normalization -----------------------------------

__global__ __launch_bounds__(256) void k_deg_init(int* __restrict__ deg, int n) {
    int i = blockIdx.x * blockDim.x + threadIdx.x;
    if (i < n) deg[i] = 1;                       // self-loop contributes 1
}

__global__ __launch_bounds__(256) void k_deg_count(int* __restrict__ deg,
                                                   const int* __restrict__ col,
                                                   int e) {
    int i = blockIdx.x * blockDim.x + threadIdx.x;
    if (i < e) atomicAdd(&deg[col[i]], 1);
}

__global__ __launch_bounds__(256) void k_dinv(const int* __restrict__ deg,
                                              float* __restrict__ dinv, int n) {
    int i = blockIdx.x * blockDim.x + threadIdx.x;
    if (i < n) {
        float d = (float)deg[i];                 // >= 1 always (self loop)
        dinv[i] = __frsqrt_rn(d);
    }
}

// ---------------- fp32 WMMA GEMM:  C[M,D] = A[M,D] @ W[D,D] -----------------
// One block = one 16-row M tile; 8 waves cover the 8 N-tiles of 16 (D=128).
// A fragment (16x4 f32): lane<16 -> M=lane, K={k,k+1}; lane>=16 -> K={k+2,k+3}
// B fragment (4x16 f32): lane<16 -> N=lane, K rows {k,k+1}; lane>=16 -> {k+2,k+3}
// C/D (16x16 f32): VGPR r -> M = r + 8*(lane/16), N = lane%16.

__global__ __launch_bounds__(256) void k_gemm_wmma(const float* __restrict__ A,
                                                   const float* __restrict__ W,
                                                   float* __restrict__ C,
                                                   int D) {
    const int lane = threadIdx.x & 31;
    const int wave = threadIdx.x >> 5;           // 0..7 -> N tile
    const int half = lane >> 4;                  // 0 or 1
    const int l16  = lane & 15;
    const int m0   = blockIdx.x * 16;
    const int n0   = wave * 16;

    const float* arow = A + (size_t)(m0 + l16) * D + 2 * half;   // 8B aligned
    const float* wcol = W + (size_t)(2 * half) * D + n0 + l16;

    v8f acc = {};
    for (int k = 0; k < D; k += 4) {
        v2f a = *(const v2f*)(arow + k);          // K = k+2h, k+2h+1
        v2f b;
        b.x = wcol[(size_t)k * D];                // row k+2h,   col n0+l16
        b.y = wcol[(size_t)(k + 1) * D];          // row k+2h+1, col n0+l16
        acc = __builtin_amdgcn_wmma_f32_16x16x4_f32(
            /*neg_a=*/false, a, /*neg_b=*/false, b,
            /*c_mod=*/(short)0, acc, /*reuse_a=*/false, /*reuse_b=*/false);
    }

    float* cb = C + (size_t)(m0 + 8 * half) * D + n0 + l16;
#pragma unroll
    for (int r = 0; r < 8; ++r) cb[(size_t)r * D] = acc[r];
}

// ---------------- seed accumulator: bias + self-loop ------------------------

__global__ __launch_bounds__(256) void k_seed(float* __restrict__ acc,
                                              const float* __restrict__ h,
                                              const float* __restrict__ dinv,
                                              const float* __restrict__ bias,
                                              int total, int D) {
    int idx = blockIdx.x * blockDim.x + threadIdx.x;
    if (idx < total) {
        int i = idx / D;
        int d = idx - i * D;
        float di = dinv[i];
        acc[idx] = bias[d] + h[idx] * di * di;
    }
}

// ---------------- edge scatter: one wave32 per edge -------------------------
// lane handles dims [4*lane, 4*lane+3] of the 128-dim message.

__global__ __launch_bounds__(256) void k_scatter(const int* __restrict__ row,
                                                 const int* __restrict__ col,
                                                 const float* __restrict__ dinv,
                                                 const float* __restrict__ h,
                                                 float* __restrict__ acc,
                                                 int E, int D) {
    int gwave = (blockIdx.x * blockDim.x + threadIdx.x) >> 5;
    int lane  = threadIdx.x & 31;
    if (gwave >= E) return;
    int r = row[gwave];
    int c = col[gwave];
    float nrm = dinv[r] * dinv[c];
    const float4 v = *(const float4*)(h + (size_t)r * D + 4 * lane);
    float* dst = acc + (size_t)c * D + 4 * lane;
    unsafeAtomicAdd(dst + 0, v.x * nrm);          // global_atomic_add_f32 in L2
    unsafeAtomicAdd(dst + 1, v.y * nrm);
    unsafeAtomicAdd(dst + 2, v.z * nrm);
    unsafeAtomicAdd(dst + 3, v.w * nrm);
}

// ---------------- ReLU in place ---------------------------------------------

__global__ __launch_bounds__(256) void k_relu(float* __restrict__ a, int total) {
    int idx = blockIdx.x * blockDim.x + threadIdx.x;
    if (idx < total) a[idx] = fmaxf(a[idx], 0.0f);
}

// ---------------------------------------------------------------------------

extern "C" void kernel_launch(void* const* d_in, const int* in_sizes, int n_in,
                              void* d_out, int out_size, void* d_ws, size_t ws_size,
                              hipStream_t stream) {
    const float* x  = (const float*)d_in[0];
    const int*   ei = (const int*)  d_in[1];
    const float* W1 = (const float*)d_in[2];
    const float* b1 = (const float*)d_in[3];
    const float* W2 = (const float*)d_in[4];
    const float* b2 = (const float*)d_in[5];

    const int D = in_sizes[3];            // 128
    const int N = in_sizes[0] / D;        // 50000
    const int E = in_sizes[1] / 2;        // 800000
    const int* row = ei;                  // edge_index[0] = sources
    const int* col = ei + E;              // edge_index[1] = targets

    // workspace layout (bytes):
    //  [0,            4N)            int   deg
    //  [4N,           8N)            float dinv
    //  [8N,           8N + 4ND)      float h    (GEMM output, both layers)
    //  [8N + 4ND,     8N + 8ND)      float a    (layer-1 accumulator)
    char*  ws   = (char*)d_ws;
    int*   deg  = (int*)ws;
    float* dinv = (float*)(ws + (size_t)4 * N);
    float* h    = (float*)(ws + (size_t)8 * N);
    float* a    = (float*)(ws + (size_t)8 * N + (size_t)4 * N * D);
    float* out  = (float*)d_out;

    const int T = 256;
    const int total = N * D;
    const int gN   = (N + T - 1) / T;
    const int gE   = (E + T - 1) / T;
    const int gTot = (total + T - 1) / T;
    const int gEdgeWaves = (E + (T / 32) - 1) / (T / 32);   // one wave32/edge
    const int gGemm = N / 16;                               // 50000/16 = 3125

    // normalization
    k_deg_init <<<gN, T, 0, stream>>>(deg, N);
    k_deg_count<<<gE, T, 0, stream>>>(deg, col, E);
    k_dinv     <<<gN, T, 0, stream>>>(deg, dinv, N);

    // ---- layer 1 ----
    k_gemm_wmma<<<gGemm, T, 0, stream>>>(x, W1, h, D);
    k_seed     <<<gTot, T, 0, stream>>>(a, h, dinv, b1, total, D);
    k_scatter  <<<gEdgeWaves, T, 0, stream>>>(row, col, dinv, h, a, E, D);
    k_relu     <<<gTot, T, 0, stream>>>(a, total);

    // ---- layer 2 ----
    k_gemm_wmma<<<gGemm, T, 0, stream>>>(a, W2, h, D);
    k_seed     <<<gTot, T, 0, stream>>>(out, h, dinv, b2, total, D);
    k_scatter  <<<gEdgeWaves, T, 0, stream>>>(row, col, dinv, h, out, E, D);
    k_relu     <<<gTot, T, 0, stream>>>(out, total);
}